// ModuleCorrelation_41910290874949
// MI455X (gfx1250) — compile-verified
//
#include <hip/hip_runtime.h>
#include <hip/hip_bf16.h>

typedef float v2f __attribute__((ext_vector_type(2)));
typedef float v8f __attribute__((ext_vector_type(8)));

#define C_    256
#define H_    192
#define W_    192
#define HW_   (H_ * W_)
#define MAXD  4
#define WIN   9
#define KSTEP 64          // 256 channels / K=4 per WMMA
#define LDSP  17          // padded LDS row stride (16 + 1)

__global__ __launch_bounds__(32)
void corr_wmma_f32_kernel(const float* __restrict__ first,
                          const float* __restrict__ second,
                          float* __restrict__ out)
{
    __shared__ float lds0[16 * LDSP];
    __shared__ float lds1[16 * LDSP];

    const int lane = threadIdx.x;          // 0..31 (wave32)
    const int wt   = blockIdx.x;           // 0..11  (w tile)
    const int h    = blockIdx.y;           // 0..191
    const int b    = blockIdx.z;           // 0..3
    const int w0   = wt * 16;

    const int m    = lane & 15;            // pixel / column index within tile
    const int cOff = (lane >> 4) << 1;     // 0 for lanes 0-15, 2 for lanes 16-31

    // ---------------- Load A (16 pixels x 256 channels) into WMMA layout ----------------
    // A VGPR pair i holds channels (4i+cOff, 4i+1+cOff) for pixel m.
    const float* aBase = first + ((b * C_) * H_ + h) * W_ + (w0 + m);
    v2f areg[KSTEP];
#pragma unroll
    for (int i = 0; i < KSTEP; ++i) {
        const float* p = aBase + (4 * i + cOff) * HW_;
        areg[i].x = p[0];
        areg[i].y = p[HW_];
    }

    const float scale = 1.0f / (float)C_;

#pragma unroll 1
    for (int dy = 0; dy < WIN; ++dy) {
        const int hs    = h + dy - MAXD;
        const int dBase = dy * WIN;

        if (hs < 0 || hs >= H_) {
            // Shifted row lies fully in zero padding -> outputs are zero.
            if (lane < 16) {
#pragma unroll
                for (int dx = 0; dx < WIN; ++dx) {
                    out[((b * (WIN * WIN) + (dBase + dx)) * H_ + h) * W_ + (w0 + m)] = 0.0f;
                }
            }
            continue;
        }

        // B tile columns: tile0 -> n = m (w0-4+m), tile1 -> n = m+8
        const int  ws0  = w0 - MAXD + m;
        const int  ws1  = ws0 + 8;
        const bool v0   = (ws0 >= 0) && (ws0 < W_);
        const bool v1   = (ws1 >= 0) && (ws1 < W_);
        const int  ws0c = min(max(ws0, 0), W_ - 1);
        const int  ws1c = min(max(ws1, 0), W_ - 1);

        const float* bRow   = second + ((b * C_) * H_ + hs) * W_;
        const float* bBase0 = bRow + ws0c;
        const float* bBase1 = bRow + ws1c;

        // Hint GL2 for the row we are about to stream (global_prefetch_b8).
        __builtin_prefetch(bBase0, 0, 1);

        v8f acc0 = {};
        v8f acc1 = {};
#pragma unroll
        for (int i = 0; i < KSTEP; ++i) {
            const int co = (4 * i + cOff) * HW_;
            // Unconditional clamped loads, then select -> v_cndmask, no EXEC change
            // (WMMA requires EXEC all ones).
            float t0a = bBase0[co];
            float t0b = bBase0[co + HW_];
            float t1a = bBase1[co];
            float t1b = bBase1[co + HW_];
            v2f b0, b1;
            b0.x = v0 ? t0a : 0.0f;
            b0.y = v0 ? t0b : 0.0f;
            b1.x = v1 ? t1a : 0.0f;
            b1.y = v1 ? t1b : 0.0f;

            acc0 = __builtin_amdgcn_wmma_f32_16x16x4_f32(
                false, areg[i], false, b0, (short)0, acc0, false, false);
            acc1 = __builtin_amdgcn_wmma_f32_16x16x4_f32(
                false, areg[i], false, b1, (short)0, acc1, false, false);
        }

        acc0 = acc0 * scale;
        acc1 = acc1 * scale;

        // ---------------- Unscramble C layout through LDS ----------------
        // C/D layout: VGPR r, lanes 0-15 -> (M=r,   N=lane)
        //                     lanes 16-31 -> (M=r+8, N=lane-16)
        const int mrow = (lane < 16) ? 0 : 8;
#pragma unroll
        for (int r = 0; r < 8; ++r) {
            lds0[(r + mrow) * LDSP + m] = acc0[r];
            lds1[(r + mrow) * LDSP + m] = acc1[r];
        }
        __syncthreads();

        if (lane < 16) {
#pragma unroll
            for (int dx = 0; dx < WIN; ++dx) {
                const int n = m + dx;                 // 0..23
                const float v = (n < 16) ? lds0[m * LDSP + n]
                                         : lds1[m * LDSP + (n - 8)];
                out[((b * (WIN * WIN) + (dBase + dx)) * H_ + h) * W_ + (w0 + m)] = v;
            }
        }
        __syncthreads();   // protect LDS reuse in next dy iteration
    }
}

extern "C" void kernel_launch(void* const* d_in, const int* in_sizes, int n_in,
                              void* d_out, int out_size, void* d_ws, size_t ws_size,
                              hipStream_t stream)
{
    const float* first  = (const float*)d_in[0];
    const float* second = (const float*)d_in[1];
    float*       out    = (float*)d_out;

    const int B = in_sizes[0] / (C_ * H_ * W_);   // = 4

    dim3 grid(W_ / 16, H_, B);
    dim3 block(32, 1, 1);
    corr_wmma_f32_kernel<<<grid, block, 0, stream>>>(first, second, out);
}